// GCN_layer_68410239091285
// MI455X (gfx1250) — compile-verified
//
#include <hip/hip_runtime.h>

#define D_FEAT 64

// ---------------------------------------------------------------------------
// SpMM, two edges per wave (defined FIRST so the disasm snippet shows it):
//   out[row[e], :] += dinv[row[e]]*dinv[col[e]] * features[col[e], :]
// Each 16-lane half-wave owns one edge; lane handles float4 (16B) of the
// 64-float feature row -> one contiguous 256B gather per edge as a single
// global_load_b128 per lane, then 4 non-returning global_atomic_add_f32.
// Prefetch the wave's next pair of feature rows (global_prefetch_b8) to hide
// the dependent col[e] -> features[col[e]] latency chain. Entire working set
// (~65MB: feat 25.6 + out 25.6 + idx 12.8 + deg 0.4) is resident in the
// 192MB L2, so gathers and the 102M scatter atomics run at L2 bandwidth;
// HBM sees only one streaming pass (~usec at 23.3 TB/s).
// ---------------------------------------------------------------------------
__global__ void gcn_spmm(const float* __restrict__ feat,
                         const int* __restrict__ row,
                         const int* __restrict__ col,
                         const float* __restrict__ dinv,
                         float* __restrict__ out, int E) {
    const int lane  = threadIdx.x & 31;
    const int sub   = lane >> 4;              // which edge of the pair
    const int l16   = lane & 15;              // lane within half-wave
    const int wave  = blockIdx.x * (blockDim.x >> 5) + (threadIdx.x >> 5);
    const int nwave = gridDim.x * (blockDim.x >> 5);
    const int estep = nwave * 2;

    for (int base = wave * 2; base < E; base += estep) {
        const int  e      = base + sub;
        const bool active = e < E;
        const int  es     = active ? e : base;   // in-bounds address for loads

        const int r = row[es];
        const int c = col[es];

        // Prefetch this wave's next pair of feature rows into cache.
        const int e2 = e + estep;
        if (e2 < E) {
            __builtin_prefetch(feat + (size_t)col[e2] * D_FEAT, 0, 3);
        }

        const float  val = dinv[r] * dinv[c];
        const float4 v   = ((const float4*)(feat + (size_t)c * D_FEAT))[l16];

        if (active) {
            float* o = out + (size_t)r * D_FEAT + (l16 << 2);
            unsafeAtomicAdd(o + 0, val * v.x);  // global_atomic_add_f32 (no rtn)
            unsafeAtomicAdd(o + 1, val * v.y);
            unsafeAtomicAdd(o + 2, val * v.z);
            unsafeAtomicAdd(o + 3, val * v.w);
        }
    }
}

// ---------------------------------------------------------------------------
// deg[row[e]] += 1.0f  (non-returning global_atomic_add_f32, 400KB target)
// ---------------------------------------------------------------------------
__global__ void gcn_degree(const int* __restrict__ row,
                           float* __restrict__ deg, int E) {
    int i = blockIdx.x * blockDim.x + threadIdx.x;
    int stride = gridDim.x * blockDim.x;
    for (; i < E; i += stride) {
        unsafeAtomicAdd(&deg[row[i]], 1.0f);
    }
}

// ---------------------------------------------------------------------------
// deg[i] = deg[i] > 0 ? rsqrt(deg[i]) : 0      (in place)
// ---------------------------------------------------------------------------
__global__ void gcn_dinv(float* __restrict__ deg, int N) {
    int i = blockIdx.x * blockDim.x + threadIdx.x;
    int stride = gridDim.x * blockDim.x;
    for (; i < N; i += stride) {
        float d = deg[i];
        deg[i] = (d > 0.f) ? rsqrtf(d) : 0.f;
    }
}

// ---------------------------------------------------------------------------
// Zero a buffer with 128-bit stores (global_store_b128).
// ---------------------------------------------------------------------------
__global__ void gcn_zero_f4(float4* __restrict__ p, long n4) {
    long i = (long)blockIdx.x * blockDim.x + threadIdx.x;
    long stride = (long)gridDim.x * blockDim.x;
    float4 z = make_float4(0.f, 0.f, 0.f, 0.f);
    for (; i < n4; i += stride) p[i] = z;
}

// ---------------------------------------------------------------------------
// Harness entry point.
//   d_in[0]: features [N*64] f32
//   d_in[1]: row      [E]    i32
//   d_in[2]: col      [E]    i32
//   d_in[3]: index    [1]    i32   (unused)
//   d_out  : [N*64] f32
//   d_ws   : >= N*4 bytes used for degree / d^-1/2 scratch
// ---------------------------------------------------------------------------
extern "C" void kernel_launch(void* const* d_in, const int* in_sizes, int n_in,
                              void* d_out, int out_size, void* d_ws, size_t ws_size,
                              hipStream_t stream) {
    const float* feat = (const float*)d_in[0];
    const int*   row  = (const int*)d_in[1];
    const int*   col  = (const int*)d_in[2];
    float*       out  = (float*)d_out;
    float*       dinv = (float*)d_ws;            // N floats of scratch

    const int N = in_sizes[0] / D_FEAT;          // 100000
    const int E = in_sizes[1];                   // 1600000

    const int TB = 256;

    // 1) Zero output and degree scratch (harness poisons them with 0xAA).
    {
        long n4 = (long)out_size / 4;            // out_size divisible by 4
        int  g  = (int)((n4 + TB - 1) / TB);
        gcn_zero_f4<<<g, TB, 0, stream>>>((float4*)out, n4);

        long d4 = (long)N / 4;                   // N divisible by 4
        int  g2 = (int)((d4 + TB - 1) / TB);
        gcn_zero_f4<<<g2, TB, 0, stream>>>((float4*)dinv, d4);
    }

    // 2) Degree histogram via f32 hw atomics.
    {
        int g = (E + TB - 1) / TB;
        if (g > 4096) g = 4096;                  // grid-stride
        gcn_degree<<<g, TB, 0, stream>>>(row, dinv, E);
    }

    // 3) dinv = rsqrt(deg) (deg > 0) else 0.
    {
        int g = (N + TB - 1) / TB;
        gcn_dinv<<<g, TB, 0, stream>>>(dinv, N);
    }

    // 4) SpMM: each wave handles 2 edges per iteration; 8192 blocks x 8
    //    waves = 65536 waves -> ~12 pair-iterations each via grid stride.
    {
        int pairs  = (E + 1) / 2;
        int blocks = (pairs + (TB / 32) - 1) / (TB / 32);
        if (blocks > 8192) blocks = 8192;
        gcn_spmm<<<blocks, TB, 0, stream>>>(feat, row, col, dinv, out, E);
    }
}